// LeakyIntegration_59854664237429
// MI455X (gfx1250) — compile-verified
//
#include <hip/hip_runtime.h>

typedef __attribute__((ext_vector_type(2))) float v2f;
typedef __attribute__((ext_vector_type(4))) float v4f;
typedef __attribute__((ext_vector_type(8))) float v8f;
typedef __attribute__((ext_vector_type(4))) int   v4i;

#define N_IN     160000
#define N_IN4    (N_IN / 4)
#define RATE     80
#define NCH      128
#define NOUT     2000
#define CHUNK    128                 // outputs per block
#define HALO     48                  // beta^48 = e^-30 ~ 9e-14 << fp32 eps
#define SREL     (CHUNK + HALO)      // 176 stage-1 partial sums per block
#define TILES    (SREL / 16)         // 11 WMMA tiles
#define XS_FLOATS (SREL * RATE)      // 14080 samples in LDS
#define XS_F4     (XS_FLOATS / 4)    // 3520 float4 loads

#if defined(__gfx1250__) && __has_builtin(__builtin_amdgcn_global_load_async_to_lds_b128)
#define HAVE_ASYNC 1
#else
#define HAVE_ASYNC 0
#endif

// alpha^e by binary exponentiation (exact-ish, ~12 roundings worst case)
__device__ __forceinline__ float powi_f(float a, int e) {
    float r = 1.0f, b = a;
    while (e) { if (e & 1) r *= b; b *= b; e >>= 1; }
    return r;
}

__global__ __launch_bounds__(256)
void LeakyIntegration_59854664237429_kernel(const float* __restrict__ x,
                                            const float* __restrict__ alphap,
                                            float* __restrict__ out) {
    __shared__ __align__(16) float xs[XS_FLOATS];  // input tile (circular slice)
    __shared__ float ss[SREL];                     // stage-1 block sums
    __shared__ float wts[RATE];                    // alpha^(79-i)
    __shared__ float bpow[HALO];                   // beta^t = alpha^(80 t)

    const int tid    = threadIdx.x;
    const int ch     = blockIdx.y;
    const int jstart = blockIdx.x * CHUNK;
    const float alpha = *alphap;

    // ---- weights ----
    if (tid < RATE) {
        wts[tid] = powi_f(alpha, RATE - 1 - tid);
    } else if (tid < RATE + HALO) {
        bpow[tid - RATE] = powi_f(alpha, RATE * (tid - RATE));
    }

    // ---- load x tile into LDS (circular indexing; float4 granules never
    //      straddle the wrap since N_IN % 4 == 0 and base % 4 == 0) ----
    int g0 = (jstart - HALO) * RATE;
    g0 %= N_IN; if (g0 < 0) g0 += N_IN;
    const int g04 = g0 >> 2;
    const v4f* xg4 = (const v4f*)(x + (size_t)ch * N_IN);
    for (int i4 = tid; i4 < XS_F4; i4 += 256) {
        int gi4 = g04 + i4;
        if (gi4 >= N_IN4) gi4 -= N_IN4;
#if HAVE_ASYNC
        __builtin_amdgcn_global_load_async_to_lds_b128(
            (__attribute__((address_space(1))) v4i*)(xg4 + gi4),
            (__attribute__((address_space(3))) v4i*)(&xs[i4 * 4]),
            0, 0);
#else
        ((v4f*)xs)[i4] = xg4[gi4];
#endif
    }
#if HAVE_ASYNC
    __builtin_amdgcn_s_wait_asynccnt(0);
#endif
    __syncthreads();

    // ---- stage 1: s[j] = sum_{i<80} alpha^(79-i) * x[80j+i] via fp32 WMMA.
    //      A[16x4]: row m = 4 consecutive samples of j-row m; B[4x16]: weight
    //      chunk broadcast across all 16 columns -> every D column equals s.
    //      A/B lane layout: lanes 0-15 carry K=0,1 (vgpr0,1), lanes 16-31 K=2,3.
    const int wave = tid >> 5;
    const int lane = tid & 31;
    const int mlo  = lane & 15;
    const int K0   = (lane >> 4) << 1;   // 0 or 2

    for (int tile = wave; tile < TILES; tile += 8) {   // wave-uniform, EXEC full
        const float* arow = &xs[(tile * 16 + mlo) * RATE + K0];
        v8f acc = {0.f, 0.f, 0.f, 0.f, 0.f, 0.f, 0.f, 0.f};
#pragma unroll
        for (int kb = 0; kb < RATE; kb += 4) {
            v2f a = *(const v2f*)(arow + kb);          // ds_load_b64 (8B aligned)
            v2f b = *(const v2f*)(&wts[kb + K0]);      // broadcast within half-wave
            acc = __builtin_amdgcn_wmma_f32_16x16x4_f32(
                false, a, false, b, (short)0, acc, false, false);
        }
        // D layout: vgpr v holds M=v (lanes 0-15) / M=v+8 (lanes 16-31),
        // replicated over N. Lanes with N==0 write the 16 sums.
        if (mlo == 0) {
            const int mbase = tile * 16 + (lane >> 4) * 8;
#pragma unroll
            for (int v = 0; v < 8; ++v) ss[mbase + v] = acc[v];
        }
    }
    __syncthreads();

    // ---- stage 2: out[j] = sum_{t<48} beta^t * s[j-t] ----
    if (tid < CHUNK) {
        const int j = jstart + tid;
        if (j < NOUT) {
            float acc = 0.0f;
#pragma unroll
            for (int t = 0; t < HALO; ++t)
                acc = fmaf(bpow[t], ss[tid + HALO - t], acc);
            out[(size_t)ch * NOUT + j] = acc;
        }
    }
}

extern "C" void kernel_launch(void* const* d_in, const int* in_sizes, int n_in,
                              void* d_out, int out_size, void* d_ws, size_t ws_size,
                              hipStream_t stream) {
    (void)in_sizes; (void)n_in; (void)out_size; (void)d_ws; (void)ws_size;
    const float* x     = (const float*)d_in[0];
    const float* alpha = (const float*)d_in[1];
    float* out = (float*)d_out;
    dim3 grid((NOUT + CHUNK - 1) / CHUNK, NCH);   // (16, 128) = 2048 blocks
    LeakyIntegration_59854664237429_kernel<<<grid, 256, 0, stream>>>(x, alpha, out);
}